// Mamba2_81544249082357
// MI455X (gfx1250) — compile-verified
//
#include <hip/hip_runtime.h>
#include <hip/hip_bf16.h>
#include <math.h>

// ---------------- problem constants (match reference) ----------------
#define DMODEL   768
#define DSTATE   64
#define DCONV    4
#define DSSM     1536
#define HEADDIM  64
#define NHEADS   24                       // DSSM / HEADDIM
#define CONVDIM  (DSSM + 2*DSTATE)        // 1664
#define DINPROJ  (2*DSSM + 2*DSTATE + NHEADS) // 3224
#define BATCH    2
#define SEQLEN   512
#define MTOK     (BATCH*SEQLEN)           // 1024

typedef float v2f __attribute__((ext_vector_type(2)));
typedef float v4f __attribute__((ext_vector_type(4)));
typedef float v8f __attribute__((ext_vector_type(8)));

// =====================================================================
// Block-cooperative GEMM:  C[M x N] = A[M x K] * B[N x K]^T
// (K contiguous in both operands; fp32 WMMA 16x16x4)
//
// Block = 256 threads (8 waves) -> 128M x 64N tile.
//   wave w: rows [w*16, w*16+16), all 64 cols; 4 v8f accumulators.
// K is processed in chunks of KC=16 staged through double-buffered LDS:
//   As[2][128][18], Bs[2][64][18]  (stride 18 floats: 8B-aligned rows,
//   bank-spread). Global loads for chunk c+1 (3x b128/thread) are issued
//   BEFORE the 16 WMMAs + ds_loads of chunk c and only consumed by the
//   ds_store at the end of the chunk, so WMMAs wait on dscnt (LDS
//   latency) instead of loadcnt (HBM/L2 latency).
//   A operand: lane%16 = M-row, lane/16 = K-pair selector
//   B operand: lane%16 = N-row, lane/16 = K-pair selector
//   C/D tile:  VGPR r, lane l -> row = r + 8*(l/16), col = l%16
// =====================================================================
#define KC    16
#define LDSS  18           // LDS row stride (floats)
#define MBLK  128
#define NBLK  64

__global__ void gemm_nt_f32_wmma(const float* __restrict__ A,
                                 const float* __restrict__ Bm,
                                 float* __restrict__ C,
                                 int M, int N, int K) {
    const int nbt = (N + NBLK - 1) / NBLK;
    const int nb  = blockIdx.x % nbt;
    const int mb  = blockIdx.x / nbt;

    const int tid  = threadIdx.x;
    const int w    = tid >> 5;              // wave 0..7
    const int lane = tid & 31;
    const int g    = lane >> 4;             // K-pair selector (0/1)
    const int r16  = lane & 15;

    __shared__ float As[2][MBLK * LDSS];
    __shared__ float Bs[2][NBLK * LDSS];

    // ---- global load mapping (per chunk) ----
    // A slab: 128 rows x 16 floats; thread -> row = tid/2, half = tid&1 (two b128)
    const int arow  = tid >> 1;
    const int ahalf = tid & 1;
    const float* agp = A + (size_t)(mb * MBLK + arow) * K + ahalf * 8;
    // B slab: 64 rows x 16 floats; thread -> row = tid/4, q = tid&3 (one b128)
    const int brow = tid >> 2;
    const int bq   = tid & 3;
    int bgrow = nb * NBLK + brow; if (bgrow >= N) bgrow = N - 1;   // tail clamp
    const float* bgp = Bm + (size_t)bgrow * K + bq * 4;

    // ---- compute-side LDS base offsets ----
    const int aoff = (w * 16 + r16) * LDSS + 2 * g;   // + ks
    const int boff = r16 * LDSS + 2 * g;              // + j*16*LDSS + ks

    v8f acc0 = {}, acc1 = {}, acc2 = {}, acc3 = {};

    // ---- prologue: stage chunk 0 ----
    {
        v4f a0 = *(const v4f*)(agp);
        v4f a1 = *(const v4f*)(agp + 4);
        v4f b0 = *(const v4f*)(bgp);
        float* as = &As[0][arow * LDSS + ahalf * 8];
        *(v2f*)(as)     = v2f{a0.x, a0.y};
        *(v2f*)(as + 2) = v2f{a0.z, a0.w};
        *(v2f*)(as + 4) = v2f{a1.x, a1.y};
        *(v2f*)(as + 6) = v2f{a1.z, a1.w};
        float* bs = &Bs[0][brow * LDSS + bq * 4];
        *(v2f*)(bs)     = v2f{b0.x, b0.y};
        *(v2f*)(bs + 2) = v2f{b0.z, b0.w};
    }
    __syncthreads();

    int buf = 0;
    for (int kk = 0; kk < K; kk += KC) {
        const bool more = (kk + KC) < K;      // uniform across block

        // ---- issue global loads for next chunk (consumed only at end) ----
        v4f a0, a1, b0;
        if (more) {
            a0 = *(const v4f*)(agp + kk + KC);
            a1 = *(const v4f*)(agp + kk + KC + 4);
            b0 = *(const v4f*)(bgp + kk + KC);
        }

        // ---- compute current chunk from LDS ----
        const float* asb = &As[buf][aoff];
        const float* bsb = &Bs[buf][boff];
        #pragma unroll
        for (int ks = 0; ks < KC; ks += 4) {
            v2f a  = *(const v2f*)(asb + ks);
            v2f q0 = *(const v2f*)(bsb +  0 * LDSS + ks);
            v2f q1 = *(const v2f*)(bsb + 16 * LDSS + ks);
            v2f q2 = *(const v2f*)(bsb + 32 * LDSS + ks);
            v2f q3 = *(const v2f*)(bsb + 48 * LDSS + ks);
            acc0 = __builtin_amdgcn_wmma_f32_16x16x4_f32(false, a, false, q0, (short)0, acc0, false, false);
            acc1 = __builtin_amdgcn_wmma_f32_16x16x4_f32(false, a, false, q1, (short)0, acc1, false, false);
            acc2 = __builtin_amdgcn_wmma_f32_16x16x4_f32(false, a, false, q2, (short)0, acc2, false, false);
            acc3 = __builtin_amdgcn_wmma_f32_16x16x4_f32(false, a, false, q3, (short)0, acc3, false, false);
        }

        // ---- store next chunk into the other buffer, flip ----
        if (more) {
            float* as = &As[buf ^ 1][arow * LDSS + ahalf * 8];
            *(v2f*)(as)     = v2f{a0.x, a0.y};
            *(v2f*)(as + 2) = v2f{a0.z, a0.w};
            *(v2f*)(as + 4) = v2f{a1.x, a1.y};
            *(v2f*)(as + 6) = v2f{a1.z, a1.w};
            float* bs = &Bs[buf ^ 1][brow * LDSS + bq * 4];
            *(v2f*)(bs)     = v2f{b0.x, b0.y};
            *(v2f*)(bs + 2) = v2f{b0.z, b0.w};
            __syncthreads();
            buf ^= 1;
        }
    }

    // ---- epilogue: write 16x64 strip ----
    const int rowbase = mb * MBLK + w * 16 + 8 * g;
    #pragma unroll
    for (int j = 0; j < 4; ++j) {
        const v8f acc = (j == 0) ? acc0 : (j == 1) ? acc1 : (j == 2) ? acc2 : acc3;
        const int col = nb * NBLK + j * 16 + r16;
        if (col < N) {
            #pragma unroll
            for (int r = 0; r < 8; ++r)
                C[(size_t)(rowbase + r) * N + col] = acc[r];
        }
    }
}

// =====================================================================
// Depthwise causal conv (k=4) over L on the xBC slice of zxbcdt,
// + bias, + SiLU.   out[b,l,c] = silu( bias[c] + sum_j w[c,j]*x[b,l-3+j,c] )
// =====================================================================
__global__ void conv_silu_kernel(const float* __restrict__ zxbcdt,
                                 const float* __restrict__ conv_w,
                                 const float* __restrict__ conv_b,
                                 float* __restrict__ xbc_out) {
    const int idx = blockIdx.x * blockDim.x + threadIdx.x;
    if (idx >= MTOK * CONVDIM) return;
    const int c  = idx % CONVDIM;
    const int bl = idx / CONVDIM;            // b*SEQLEN + l
    const int l  = bl % SEQLEN;

    float acc = conv_b[c];
    #pragma unroll
    for (int j = 0; j < DCONV; ++j) {
        const int ll = l - (DCONV - 1) + j;
        if (ll >= 0) {
            const float xv = zxbcdt[(size_t)(bl - (DCONV - 1) + j) * DINPROJ + DSSM + c];
            acc = fmaf(xv, conv_w[c * DCONV + j], acc);
        }
    }
    const float s = acc / (1.0f + __expf(-acc));   // silu
    xbc_out[(size_t)bl * CONVDIM + c] = s;
}

// =====================================================================
// SSM recurrent scan. One workgroup per (b,h). 256 threads.
// thread t: p = t/4 (HEADDIM row), nc = t%4 (16-wide D_STATE chunk).
// State (64x64 fp32) lives in VGPRs: 16 floats/thread.
// x/B/C staged through LDS each step; y reduced with wave32 shfl_xor.
// =====================================================================
__global__ void ssm_scan_kernel(const float* __restrict__ zxbcdt,
                                const float* __restrict__ xbc,
                                const float* __restrict__ dt_bias,
                                const float* __restrict__ A_log,
                                const float* __restrict__ D_param,
                                float* __restrict__ y) {
    const int bh = blockIdx.x;               // 0 .. BATCH*NHEADS-1
    const int b  = bh / NHEADS;
    const int h  = bh % NHEADS;
    const int t  = threadIdx.x;              // 0..255
    const int p  = t >> 2;
    const int nc = t & 3;

    const float Ah  = -__expf(A_log[h]);
    const float dtb = dt_bias[h];
    const float Dp  = D_param[h];

    __shared__ float sx[HEADDIM];
    __shared__ float sB[DSTATE];
    __shared__ float sC[DSTATE];

    float s[16];
    #pragma unroll
    for (int i = 0; i < 16; ++i) s[i] = 0.0f;

    for (int l = 0; l < SEQLEN; ++l) {
        const size_t rowc = (size_t)(b * SEQLEN + l) * CONVDIM;
        const size_t rowz = (size_t)(b * SEQLEN + l) * DINPROJ;

        if (t < 64)        sx[t]       = xbc[rowc + h * HEADDIM + t];
        else if (t < 128)  sB[t - 64]  = xbc[rowc + DSSM + (t - 64)];
        else if (t < 192)  sC[t - 128] = xbc[rowc + DSSM + DSTATE + (t - 128)];
        __syncthreads();

        // dt = clip(softplus(dt_raw + dt_bias), 1e-6, 1000)
        const float dv  = zxbcdt[rowz + (DSSM + CONVDIM) + h] + dtb;
        float dt = (dv > 20.0f) ? dv : __logf(1.0f + __expf(dv));
        dt = fminf(fmaxf(dt, 1e-6f), 1000.0f);
        const float dA = fminf(fmaxf(__expf(dt * Ah), 1e-6f), 1000.0f);

        const float xv  = sx[p];
        const float dtx = dt * xv;

        float ysum = 0.0f;
        #pragma unroll
        for (int i = 0; i < 16; ++i) {
            const int n = nc * 16 + i;
            float sv = fmaf(s[i], dA, dtx * sB[n]);
            sv = fminf(fmaxf(sv, -1000.0f), 1000.0f);
            s[i] = sv;
            ysum = fmaf(sv, sC[n], ysum);
        }
        // reduce across the 4 nc lanes (same wave32)
        ysum += __shfl_xor(ysum, 1);
        ysum += __shfl_xor(ysum, 2);
        if (nc == 0)
            y[(size_t)(b * SEQLEN + l) * DSSM + h * HEADDIM + p] = fmaf(Dp, xv, ysum);
        __syncthreads();
    }
}

// =====================================================================
// y *= silu(z)   (z = first DSSM columns of zxbcdt)
// =====================================================================
__global__ void gate_kernel(const float* __restrict__ zxbcdt,
                            float* __restrict__ y) {
    const int idx = blockIdx.x * blockDim.x + threadIdx.x;
    if (idx >= MTOK * DSSM) return;
    const int row = idx / DSSM;
    const int c   = idx % DSSM;
    const float z  = zxbcdt[(size_t)row * DINPROJ + c];
    const float sz = z / (1.0f + __expf(-z));
    y[idx] *= sz;
}

// =====================================================================
// launch
// =====================================================================
extern "C" void kernel_launch(void* const* d_in, const int* in_sizes, int n_in,
                              void* d_out, int out_size, void* d_ws, size_t ws_size,
                              hipStream_t stream) {
    const float* u       = (const float*)d_in[0];   // (B,L,DMODEL)
    const float* W_in    = (const float*)d_in[1];   // (DINPROJ, DMODEL)
    const float* conv_w  = (const float*)d_in[2];   // (CONVDIM, DCONV)
    const float* conv_b  = (const float*)d_in[3];   // (CONVDIM,)
    const float* dt_bias = (const float*)d_in[4];   // (NHEADS,)
    const float* A_log   = (const float*)d_in[5];   // (NHEADS,)
    const float* D_param = (const float*)d_in[6];   // (NHEADS,)
    const float* W_out   = (const float*)d_in[7];   // (DMODEL, DSSM)
    float* out = (float*)d_out;                     // (B,L,DMODEL)

    float* ws      = (float*)d_ws;
    float* zxbcdt  = ws;                                   // MTOK * DINPROJ
    float* xbc     = zxbcdt + (size_t)MTOK * DINPROJ;      // MTOK * CONVDIM
    float* ybuf    = xbc    + (size_t)MTOK * CONVDIM;      // MTOK * DSSM

    // ---- 1) in-projection: zxbcdt = u @ W_in^T ----
    {
        const int mbt = MTOK / MBLK;                  // 8
        const int nbt = (DINPROJ + NBLK - 1) / NBLK;  // 51
        gemm_nt_f32_wmma<<<mbt * nbt, 256, 0, stream>>>(u, W_in, zxbcdt,
                                                        MTOK, DINPROJ, DMODEL);
    }

    // ---- 2) depthwise conv + bias + SiLU ----
    {
        const int n = MTOK * CONVDIM;
        conv_silu_kernel<<<(n + 255) / 256, 256, 0, stream>>>(zxbcdt, conv_w, conv_b, xbc);
    }

    // ---- 3) SSM scan ----
    ssm_scan_kernel<<<BATCH * NHEADS, 256, 0, stream>>>(zxbcdt, xbc, dt_bias,
                                                        A_log, D_param, ybuf);

    // ---- 4) gating: y *= silu(z) ----
    {
        const int n = MTOK * DSSM;
        gate_kernel<<<(n + 255) / 256, 256, 0, stream>>>(zxbcdt, ybuf);
    }

    // ---- 5) out-projection: out = y @ W_out^T ----
    {
        const int mbt = MTOK / MBLK;                  // 8
        const int nbt = (DMODEL + NBLK - 1) / NBLK;   // 12
        gemm_nt_f32_wmma<<<mbt * nbt, 256, 0, stream>>>(ybuf, W_out, out,
                                                        MTOK, DMODEL, DSSM);
    }
}